// LAL_10514079941286
// MI455X (gfx1250) — compile-verified
//
#include <hip/hip_runtime.h>

typedef __attribute__((ext_vector_type(2))) float v2f;
typedef __attribute__((ext_vector_type(8))) float v8f;
typedef __attribute__((ext_vector_type(4))) unsigned int u32x4;
typedef __attribute__((ext_vector_type(4))) int i32x4;
typedef __attribute__((ext_vector_type(8))) int i32x8;

#define B_  8
#define N_  1024
#define C_  512
#define K_  20
#define M_  64
#define R_  (B_*N_)      /* 8192 rows */
#define NCOLS1 128       /* Y(64) | Z(64) */
#define BN_EPS 1e-5f
#define NEG_SLOPE 0.2f

// d_ws float offsets
#define OFF_BP1   0u          // 65536  : repacked [W1a | W1b-W1a]^T, pair-interleaved
#define OFF_BP2   65536u      // 32768  : repacked W2^T, pair-interleaved
#define OFF_YZ    98304u      // 1048576: Y|Z per point (8192 x 128)
#define OFF_HMAX  1146880u    // 524288 : max_k h (8192 x 64), pre-BN
#define OFF_STATS 1671168u    // 1152   : s1sum(64) s1sq(64) s2sum(512) s2sq(512)
#define OFF_S1    1672320u    // 64
#define OFF_B1    1672384u    // 64
#define OFF_S2    1672448u    // 512
#define OFF_B2    1672960u    // 512

static __device__ __forceinline__ float leaky(float v) { return v >= 0.f ? v : NEG_SLOPE * v; }

static __device__ __forceinline__ v8f wmma4(v2f a, v2f b, v8f c) {
  // V_WMMA_F32_16X16X4_F32: D = A(16x4,f32) x B(4x16,f32) + C(16x16,f32)
  return __builtin_amdgcn_wmma_f32_16x16x4_f32(
      /*neg_a=*/false, a, /*neg_b=*/false, b,
      /*c_mod=*/(short)0, c, /*reuse_a=*/false, /*reuse_b=*/false);
}

// ---- Tensor Data Mover: 2D tile load (8-byte elements) per CDNA5 ISA
#if __clang_major__ >= 23
#define TDM_ISSUE(g0, g1) \
  __builtin_amdgcn_tensor_load_to_lds((g0), (g1), (i32x4)0, (i32x4)0, (i32x8)0, 0)
#else
#define TDM_ISSUE(g0, g1) \
  __builtin_amdgcn_tensor_load_to_lds((g0), (g1), (i32x4)0, (i32x4)0, 0)
#endif

static __device__ __forceinline__ void tdm_load_2d(unsigned lds_off, const void* gptr,
                                                   unsigned tensor_d0, unsigned tensor_d1,
                                                   unsigned tile_d0, unsigned tile_d1,
                                                   unsigned long long stride0_elems) {
  unsigned long long ga = (unsigned long long)(uintptr_t)gptr;
  u32x4 g0;
  g0[0] = 1u;                                              // count=1, user descriptor
  g0[1] = lds_off;                                         // lds_addr (bytes)
  g0[2] = (unsigned)(ga & 0xffffffffu);                    // global_addr[31:0]
  g0[3] = (unsigned)((ga >> 32) & 0x1ffffffu) | 0x80000000u; // global_addr[56:32] | type=2
  i32x8 g1;
  g1[0] = (int)(3u << 16);                                 // data_size=3 (8B), no multicast
  g1[1] = (int)((tensor_d0 & 0xffffu) << 16);              // tensor_dim0[15:0] @ bits 63:48
  g1[2] = (int)((tensor_d0 >> 16) | ((tensor_d1 & 0xffffu) << 16));
  g1[3] = (int)((tensor_d1 >> 16) | (tile_d0 << 16));      // tile_dim0 @ bits 127:112
  g1[4] = (int)(tile_d1 & 0xffffu);                        // tile_dim1; tile_dim2=0
  g1[5] = (int)(stride0_elems & 0xffffffffu);              // tensor_dim0_stride[31:0]
  g1[6] = (int)((stride0_elems >> 32) & 0xffffu);          // stride[47:32]; dim1_stride=0
  g1[7] = 0;
  TDM_ISSUE(g0, g1);
}

// ---------------------------------------------------------------- zero stats
__global__ void k_zero(float* __restrict__ stats) {
  int i = blockIdx.x * blockDim.x + threadIdx.x;
  if (i < 1152) stats[i] = 0.f;
}

// ------------------------------------------- repack W1 -> [W1a | W1b-W1a]^T
// bp1[((k>>1)*128 + n)*2 + (k&1)] = Bmat[k][n], k in [0,512), n in [0,128)
__global__ void k_prep_w1(const float* __restrict__ W1, float* __restrict__ bp1) {
  int t = blockIdx.x * 256 + threadIdx.x;   // 32768 k-pairs x n
  int kp = t >> 7, n = t & 127;
  int k = kp * 2;
  float v0, v1;
  if (n < 64) {                 // Y columns: W1a[n][k]
    v0 = W1[n * 1024 + k];
    v1 = W1[n * 1024 + k + 1];
  } else {                      // Z columns: (W1b - W1a)[m][k]
    int m = n - 64;
    v0 = W1[m * 1024 + 512 + k]     - W1[m * 1024 + k];
    v1 = W1[m * 1024 + 512 + k + 1] - W1[m * 1024 + k + 1];
  }
  bp1[t * 2]     = v0;
  bp1[t * 2 + 1] = v1;
}

// ------------------------------------------------------- repack W2 -> W2^T
// bp2[((m>>1)*512 + d)*2 + (m&1)] = W2[d][m]
__global__ void k_prep_w2(const float* __restrict__ W2, float* __restrict__ bp2) {
  int t = blockIdx.x * 256 + threadIdx.x;   // 16384 m-pairs x d
  int mp = t >> 9, d = t & 511;
  int m = mp * 2;
  bp2[t * 2]     = W2[d * 64 + m];
  bp2[t * 2 + 1] = W2[d * 64 + m + 1];
}

// -------------------------------------------- GEMM1: YZ = x @ Bmat (8192x512 x 512x128)
// Weights staged by TDM with double-buffered LDS; WMMA overlaps the DMA.
__global__ void __launch_bounds__(256) k_gemm1(const float* __restrict__ x,
                                               const float* __restrict__ bp1,
                                               float* __restrict__ YZ) {
  __shared__ float ldsB[2][64 * 128];       // 2 x 32 KB weight chunks, pair-interleaved
  const int wave = threadIdx.x >> 5;
  const int lane = threadIdx.x & 31;
  const int half = lane >> 4;
  const int l16  = lane & 15;
  const int row0 = blockIdx.x * 128 + wave * 16;
  const float* xrow = x + (row0 + l16) * C_;
  v8f acc[8] = {};

  if (wave == 0) {   // prologue: DMA chunk 0 into buffer 0 (contiguous 4096 x 8B)
    tdm_load_2d((unsigned)(uintptr_t)&ldsB[0][0], bp1, 4096u, 1u, 4096u, 1u, 4096ull);
  }
  for (int c = 0; c < 8; ++c) {
    if (wave == 0) {
      if (c + 1 < 8) {  // DMA next chunk into the other buffer while we compute
        tdm_load_2d((unsigned)(uintptr_t)&ldsB[(c + 1) & 1][0],
                    bp1 + (c + 1) * 8192, 4096u, 1u, 4096u, 1u, 4096ull);
        __builtin_amdgcn_s_wait_tensorcnt(1);  // chunk c landed; c+1 still in flight
      } else {
        __builtin_amdgcn_s_wait_tensorcnt(0);  // last chunk landed
      }
    }
    __syncthreads();                        // publish buffer c&1 to all waves
    const v2f* ldsB2 = (const v2f*)&ldsB[c & 1][0];
    const int kc = c * 64;
    for (int k0 = 0; k0 < 64; k0 += 4) {
      v2f a = *(const v2f*)(xrow + kc + k0 + 2 * half);   // A frag: K pair per half-wave
      const v2f* brow = ldsB2 + (k0 / 2 + half) * 128 + l16;
#pragma unroll
      for (int t = 0; t < 8; ++t) {
        v2f b = brow[t * 16];
        acc[t] = wmma4(a, b, acc[t]);
      }
    }
    __syncthreads();                        // all reads of buffer c&1 done before rewrite
  }
#pragma unroll
  for (int t = 0; t < 8; ++t)
#pragma unroll
    for (int g = 0; g < 8; ++g)
      YZ[(row0 + g + 8 * half) * NCOLS1 + t * 16 + l16] = acc[t][g];
}

// ---------------- gather + bias + max over K + BN1 statistics accumulation
__global__ void __launch_bounds__(256) k_gather(const float* __restrict__ YZ,
                                                const int* __restrict__ idx,
                                                float* __restrict__ hmax,
                                                float* __restrict__ stats) {
  const int wave = threadIdx.x >> 5, lane = threadIdx.x & 31;
  const int w = blockIdx.x * 8 + wave;      // 1024 waves total
  const v2f* YZ2 = (const v2f*)YZ;          // 64 float2 per row
  v2f* hm2 = (v2f*)hmax;                    // 32 float2 per row
  float s0 = 0.f, s1 = 0.f, q0 = 0.f, q1 = 0.f;
  for (int p = w; p < R_; p += 1024) {
    int b = p >> 10;
    v2f z = YZ2[p * 64 + 32 + lane];        // Z part (cols 64..127)
    float m0 = -3.4e38f, m1 = -3.4e38f;
    for (int k = 0; k < K_; ++k) {
      int j = idx[p * K_ + k];
      v2f y = YZ2[(b * N_ + j) * 64 + lane];  // Y part of neighbor
      float h0 = y[0] + z[0], h1 = y[1] + z[1];
      m0 = fmaxf(m0, h0); m1 = fmaxf(m1, h1);
      s0 += h0; q0 += h0 * h0; s1 += h1; q1 += h1 * h1;
    }
    v2f mv; mv[0] = m0; mv[1] = m1;
    hm2[p * 32 + lane] = mv;
  }
  atomicAdd(stats +      2 * lane,     s0);
  atomicAdd(stats +      2 * lane + 1, s1);
  atomicAdd(stats + 64 + 2 * lane,     q0);
  atomicAdd(stats + 64 + 2 * lane + 1, q1);
}

// ------------------------------------------------------------ BN1 finalize
__global__ void k_fin1(const float* __restrict__ stats,
                       const float* __restrict__ gamma1, const float* __restrict__ beta1,
                       float* __restrict__ s1, float* __restrict__ b1) {
  int m = threadIdx.x;
  if (m >= 64) return;
  const float inv = 1.f / (float)(R_ * K_);
  float mean = stats[m] * inv;
  float var  = stats[64 + m] * inv - mean * mean;
  float s = gamma1[m] * rsqrtf(var + BN_EPS);
  s1[m] = s;
  b1[m] = beta1[m] - mean * s;
}

// ------- GEMM2: out = leaky(BN1(hmax)) @ W2^T  (8192x64 x 64x512), pre-BN2
// 2-D TDM tile load: 32 rows x 128 x 8B, row stride 512 elements.
__global__ void __launch_bounds__(256) k_gemm2(const float* __restrict__ hmax,
                                               const float* __restrict__ bp2,
                                               const float* __restrict__ s1,
                                               const float* __restrict__ b1,
                                               float* __restrict__ out) {
  __shared__ float ldsB[64 * 128];          // 32 KB: K=64 x 128-col slice of W2^T
  const int wave = threadIdx.x >> 5, lane = threadIdx.x & 31;
  const int half = lane >> 4, l16 = lane & 15;
  const int row0 = blockIdx.x * 128 + wave * 16;
  const int d0   = blockIdx.y * 128;
  if (wave == 0) {
    tdm_load_2d((unsigned)(uintptr_t)&ldsB[0], (const v2f*)bp2 + d0,
                512u, 32u, 128u, 32u, 512ull);
    __builtin_amdgcn_s_wait_tensorcnt(0);
  }
  __syncthreads();
  const v2f* ldsB2 = (const v2f*)ldsB;
  const float* hrow = hmax + (row0 + l16) * 64;
  v8f acc[8] = {};
  for (int k0 = 0; k0 < 64; k0 += 4) {
    int kk = k0 + 2 * half;
    v2f h  = *(const v2f*)(hrow + kk);
    v2f sv = *(const v2f*)(s1 + kk);
    v2f bv = *(const v2f*)(b1 + kk);
    v2f a;                                   // fused BN1 + leaky on the A fragment
    a[0] = leaky(sv[0] * h[0] + bv[0]);
    a[1] = leaky(sv[1] * h[1] + bv[1]);
    const v2f* brow = ldsB2 + (k0 / 2 + half) * 128 + l16;
#pragma unroll
    for (int t = 0; t < 8; ++t) {
      v2f b = brow[t * 16];
      acc[t] = wmma4(a, b, acc[t]);
    }
  }
#pragma unroll
  for (int t = 0; t < 8; ++t)
#pragma unroll
    for (int g = 0; g < 8; ++g)
      out[(row0 + g + 8 * half) * C_ + d0 + t * 16 + l16] = acc[t][g];
}

// ------------------------------------------------------------- BN2 stats
__global__ void __launch_bounds__(256) k_stats2(const float* __restrict__ out,
                                                float* __restrict__ stats) {
  const int r0 = blockIdx.x * 64;
  const int c0 = threadIdx.x, c1 = threadIdx.x + 256;
  float a0 = 0.f, a1 = 0.f, q0 = 0.f, q1 = 0.f;
  for (int r = r0; r < r0 + 64; ++r) {
    float v0 = out[r * C_ + c0], v1 = out[r * C_ + c1];
    a0 += v0; q0 += v0 * v0; a1 += v1; q1 += v1 * v1;
  }
  atomicAdd(stats + 128 + c0, a0); atomicAdd(stats + 640 + c0, q0);
  atomicAdd(stats + 128 + c1, a1); atomicAdd(stats + 640 + c1, q1);
}

// ------------------------------------------------------------ BN2 finalize
__global__ void k_fin2(const float* __restrict__ stats,
                       const float* __restrict__ gamma2, const float* __restrict__ beta2,
                       float* __restrict__ s2, float* __restrict__ b2) {
  int d = threadIdx.x;
  const float inv = 1.f / (float)R_;
  float mean = stats[128 + d] * inv;
  float var  = stats[640 + d] * inv - mean * mean;
  float s = gamma2[d] * rsqrtf(var + BN_EPS);
  s2[d] = s;
  b2[d] = beta2[d] - mean * s;
}

// --------------------------------------- in-place BN2 + leaky on d_out
__global__ void __launch_bounds__(256) k_norm(float* __restrict__ out,
                                              const float* __restrict__ s2,
                                              const float* __restrict__ b2) {
  int tid = blockIdx.x * 256 + threadIdx.x;  // 1,048,576 threads x float4
  float4* o4 = (float4*)out;
  float4 v = o4[tid];
  int d = (tid * 4) & 511;
  const float4 s = *(const float4*)(s2 + d);
  const float4 b = *(const float4*)(b2 + d);
  v.x = leaky(s.x * v.x + b.x);
  v.y = leaky(s.y * v.y + b.y);
  v.z = leaky(s.z * v.z + b.z);
  v.w = leaky(s.w * v.w + b.w);
  o4[tid] = v;
}

extern "C" void kernel_launch(void* const* d_in, const int* in_sizes, int n_in,
                              void* d_out, int out_size, void* d_ws, size_t ws_size,
                              hipStream_t stream) {
  const float* x   = (const float*)d_in[0];
  const int*   idx = (const int*)  d_in[1];
  const float* W1  = (const float*)d_in[2];
  const float* g1  = (const float*)d_in[3];
  const float* be1 = (const float*)d_in[4];
  const float* W2  = (const float*)d_in[5];
  const float* g2  = (const float*)d_in[6];
  const float* be2 = (const float*)d_in[7];
  float* out = (float*)d_out;
  float* ws  = (float*)d_ws;

  float* bp1   = ws + OFF_BP1;
  float* bp2   = ws + OFF_BP2;
  float* YZ    = ws + OFF_YZ;
  float* hmax  = ws + OFF_HMAX;
  float* stats = ws + OFF_STATS;
  float* s1    = ws + OFF_S1;
  float* b1    = ws + OFF_B1;
  float* s2    = ws + OFF_S2;
  float* b2    = ws + OFF_B2;

  k_zero   <<<5,    256, 0, stream>>>(stats);
  k_prep_w1<<<128,  256, 0, stream>>>(W1, bp1);
  k_prep_w2<<<64,   256, 0, stream>>>(W2, bp2);
  k_gemm1  <<<64,   256, 0, stream>>>(x, bp1, YZ);
  k_gather <<<128,  256, 0, stream>>>(YZ, idx, hmax, stats);
  k_fin1   <<<1,    64,  0, stream>>>(stats, g1, be1, s1, b1);
  k_gemm2  <<<dim3(64, 4), 256, 0, stream>>>(hmax, bp2, s1, b1, out);
  k_stats2 <<<128,  256, 0, stream>>>(out, stats);
  k_fin2   <<<1,    512, 0, stream>>>(stats, g2, be2, s2, b2);
  k_norm   <<<4096, 256, 0, stream>>>(out, s2, b2);
}